// QRNN_60498909332052
// MI455X (gfx1250) — compile-verified
//
#include <hip/hip_runtime.h>

// ---------------- problem constants (from reference) ----------------
#define B_   8
#define T_   4096
#define D_   256                 // Din == Dout
#define NCH  16                  // scan chunks per (b,d) channel
#define CH   (T_ / NCH)          // 256 steps per chunk
#define BTD  (B_ * T_ * D_)      // 8,388,608 floats per gate

// LDS tile geometry for the gate GEMM
#define XROWS 129                // 128 t-rows + 1 history row for tap 0
#define XSTR  260                // padded row stride (floats): banks = (4*row + k) % 64, conflict-free
#define WELEMS (3 * 2 * D_ * 16) // 3 gates x 2 taps x 256 K x 16 N

typedef __attribute__((ext_vector_type(2))) float v2f;
typedef __attribute__((ext_vector_type(4))) float v4f;
typedef __attribute__((ext_vector_type(8))) float v8f;
typedef __attribute__((ext_vector_type(4))) int   v4i;

#ifndef __has_builtin
#define __has_builtin(x) 0
#endif

// gfx1250 async global->LDS staging (ASYNCcnt-tracked, no VGPR round trip).
#if __has_builtin(__builtin_amdgcn_global_load_async_to_lds_b128)
#define HAVE_ASYNC_LDS 1
#else
#define HAVE_ASYNC_LDS 0
#endif

#define GLOBAL_AS __attribute__((address_space(1)))
#define LDS_AS    __attribute__((address_space(3)))

#if HAVE_ASYNC_LDS
__device__ __forceinline__ void async_copy16(const float* src, float* lds_dst) {
  __builtin_amdgcn_global_load_async_to_lds_b128(
      (GLOBAL_AS v4i*)const_cast<float*>(src),
      (LDS_AS v4i*)lds_dst,
      /*imm offset*/ 0, /*cpol*/ 0);
}
__device__ __forceinline__ void wait_asynccnt0() {
#if __has_builtin(__builtin_amdgcn_s_wait_asynccnt)
  __builtin_amdgcn_s_wait_asynccnt(0);
#else
  asm volatile("s_wait_asynccnt 0x0" ::: "memory");
#endif
}
#endif

__device__ __forceinline__ float fast_sigmoid(float x) {
  // sigmoid via v_exp_f32 + v_rcp_f32
  float e = __builtin_amdgcn_exp2f(-x * 1.4426950408889634f);
  return __builtin_amdgcn_rcpf(1.0f + e);
}
__device__ __forceinline__ float fast_tanh(float x) {
  return 2.0f * fast_sigmoid(2.0f * x) - 1.0f;
}

// =====================================================================
// Kernel 1: causal window conv for all three gates via f32 WMMA.
// Block = 256 threads (8 wave32). Each block: 128 t-rows x 16 dout cols,
// all 3 gates. grid = (B*T/128, D/16).
// =====================================================================
__global__ void __launch_bounds__(256)
qrnn_gates_wmma(const float* __restrict__ x,
                const float* __restrict__ Wz,
                const float* __restrict__ Wf,
                const float* __restrict__ Wo,
                const float* __restrict__ bz,
                const float* __restrict__ bf,
                const float* __restrict__ bo,
                float* __restrict__ gz,
                float* __restrict__ gf,
                float* __restrict__ go)
{
  extern __shared__ __align__(16) float smem[];
  float* lds_x = smem;                   // [XROWS][XSTR]
  float* lds_w = smem + XROWS * XSTR;    // [3][2][256][16]

  const int tid    = threadIdx.x;
  const int bb     = blockIdx.x >> 5;          // batch index (T/128 == 32 tiles per batch)
  const int t0base = (blockIdx.x & 31) * 128;  // t offset within batch
  const int n0     = blockIdx.y * 16;          // dout tile

  // ---- stage x rows [t0base-1 .. t0base+127] into LDS (row 0 zeroed at batch start)
  const float* xb = x + (long)bb * T_ * D_;
  for (int i = tid; i < XROWS * (D_ / 4); i += 256) {
    int r  = i >> 6;          // row 0..128
    int c4 = (i & 63) << 2;   // column (float4 granules)
    float* dst = lds_x + r * XSTR + c4;
    if (t0base == 0 && r == 0) {
      *(v4f*)dst = (v4f)0.0f;
    } else {
      const float* src = xb + (long)(t0base - 1 + r) * D_ + c4;
#if HAVE_ASYNC_LDS
      async_copy16(src, dst);
#else
      *(v4f*)dst = *(const v4f*)src;
#endif
    }
  }
  // ---- stage weight slices for all 3 gates, both taps, 16 output cols
  for (int i = tid; i < WELEMS / 4; i += 256) {
    int g    = i / 2048;          // gate
    int rem  = i - g * 2048;
    int tap  = rem >> 10;
    int rem2 = rem & 1023;
    int kk   = rem2 >> 2;
    int nn4  = (rem2 & 3) << 2;
    const float* Wsrc = (g == 0) ? Wz : (g == 1) ? Wf : Wo;
    const float* src  = Wsrc + tap * (D_ * D_) + kk * D_ + n0 + nn4;
    float* dst = lds_w + ((g * 2 + tap) * D_ + kk) * 16 + nn4;
#if HAVE_ASYNC_LDS
    async_copy16(src, dst);
#else
    *(v4f*)dst = *(const v4f*)src;
#endif
  }
#if HAVE_ASYNC_LDS
  wait_asynccnt0();
#endif
  __syncthreads();

  const int lane = tid & 31;
  const int wv   = tid >> 5;     // wave id -> 16-row sub-tile
  const int nl   = lane & 15;    // A: row m; B/C/D: column n
  const int kh   = lane >> 4;    // K-half selector per ISA fragment layout

  const float biasz = bz[n0 + nl];
  const float biasf = bf[n0 + nl];
  const float biaso = bo[n0 + nl];
  v8f accz, accf, acco;
#pragma unroll
  for (int r = 0; r < 8; ++r) { accz[r] = biasz; accf[r] = biasf; acco[r] = biaso; }

#pragma unroll
  for (int tap = 0; tap < 2; ++tap) {
    const float* arow = lds_x + (wv * 16 + nl + tap) * XSTR;    // row t0+m+tap-1 (+1 history row)
    const float* wb0  = lds_w + ((0 * 2 + tap) * D_) * 16;
    const float* wb1  = lds_w + ((1 * 2 + tap) * D_) * 16;
    const float* wb2  = lds_w + ((2 * 2 + tap) * D_) * 16;
#pragma unroll 4
    for (int k0 = 0; k0 < D_; k0 += 4) {
      const int k = k0 + 2 * kh;
      v2f a = *(const v2f*)(arow + k);        // A frag: {K=k, K=k+1} for this lane half
      v2f bz2, bf2, bo2;                      // B frag: vgpr0 row k, vgpr1 row k+1
      bz2.x = wb0[(k    ) * 16 + nl];
      bz2.y = wb0[(k + 1) * 16 + nl];
      bf2.x = wb1[(k    ) * 16 + nl];
      bf2.y = wb1[(k + 1) * 16 + nl];
      bo2.x = wb2[(k    ) * 16 + nl];
      bo2.y = wb2[(k + 1) * 16 + nl];
      accz = __builtin_amdgcn_wmma_f32_16x16x4_f32(false, a, false, bz2, (short)0, accz, false, false);
      accf = __builtin_amdgcn_wmma_f32_16x16x4_f32(false, a, false, bf2, (short)0, accf, false, false);
      acco = __builtin_amdgcn_wmma_f32_16x16x4_f32(false, a, false, bo2, (short)0, acco, false, false);
    }
  }

  // ---- activations + store (C/D layout: VGPR r -> M = r + 8*kh, N = nl)
  const long rowbase = (long)bb * T_ + t0base + wv * 16;
#pragma unroll
  for (int r = 0; r < 8; ++r) {
    const long idx = (rowbase + r + 8 * kh) * D_ + n0 + nl;
    gz[idx] = fast_tanh(accz[r]);
    gf[idx] = fast_sigmoid(accf[r]);
    go[idx] = fast_sigmoid(acco[r]);
  }
}

// =====================================================================
// Kernel 2: per-chunk affine reduction of the recurrence.
// c_t = f*c + (1-f)*z  ==>  over a chunk: c_out = A*c_in + B.
// grid = B*NCH blocks of 256 threads (thread = one d channel). Coalesced.
// =====================================================================
__global__ void __launch_bounds__(256)
qrnn_chunk_reduce(const float* __restrict__ gz,
                  const float* __restrict__ gf,
                  float* __restrict__ chA,
                  float* __restrict__ chB)
{
  const int d  = threadIdx.x;
  const int b  = blockIdx.x >> 4;
  const int ch = blockIdx.x & 15;
  const long base = ((long)b * T_ + (long)ch * CH) * D_ + d;
  float A = 1.0f, Bc = 0.0f;
#pragma unroll 8
  for (int t = 0; t < CH; ++t) {
    const float f = gf[base + (long)t * D_];
    const float z = gz[base + (long)t * D_];
    __builtin_prefetch(&gf[base + (long)(t + 32) * D_], 0, 0);
    A  = A * f;
    Bc = f * Bc + (1.0f - f) * z;
  }
  const long ci = (long)blockIdx.x * D_ + d;
  chA[ci] = A;
  chB[ci] = Bc;
}

// =====================================================================
// Kernel 3: fold prior-chunk affine maps for the carry-in, replay the
// chunk, emit h = o * c. Same grid/coalescing as kernel 2.
// =====================================================================
__global__ void __launch_bounds__(256)
qrnn_chunk_apply(const float* __restrict__ gz,
                 const float* __restrict__ gf,
                 const float* __restrict__ go,
                 const float* __restrict__ chA,
                 const float* __restrict__ chB,
                 float* __restrict__ h)
{
  const int d  = threadIdx.x;
  const int b  = blockIdx.x >> 4;
  const int ch = blockIdx.x & 15;

  // carry-in: compose chunks 0..ch-1 of this channel
  float c = 0.0f;
  const long abase = (long)b * NCH * D_ + d;
  for (int pc = 0; pc < ch; ++pc) {
    const float A  = chA[abase + (long)pc * D_];
    const float Bv = chB[abase + (long)pc * D_];
    c = A * c + Bv;
  }

  const long base = ((long)b * T_ + (long)ch * CH) * D_ + d;
#pragma unroll 8
  for (int t = 0; t < CH; ++t) {
    const long i = base + (long)t * D_;
    const float f = gf[i];
    const float z = gz[i];
    const float o = go[i];
    __builtin_prefetch(&go[i + (long)32 * D_], 0, 0);
    c = f * c + (1.0f - f) * z;
    h[i] = o * c;
  }
}

// =====================================================================
extern "C" void kernel_launch(void* const* d_in, const int* in_sizes, int n_in,
                              void* d_out, int out_size, void* d_ws, size_t ws_size,
                              hipStream_t stream) {
  (void)in_sizes; (void)n_in; (void)out_size; (void)ws_size;
  const float* x  = (const float*)d_in[0];
  const float* Wz = (const float*)d_in[1];
  const float* Wf = (const float*)d_in[2];
  const float* Wo = (const float*)d_in[3];
  const float* bz = (const float*)d_in[4];
  const float* bf = (const float*)d_in[5];
  const float* bo = (const float*)d_in[6];

  float* gz  = (float*)d_ws;                 // [B,T,D] tanh(z)
  float* gf  = gz + (size_t)BTD;             // [B,T,D] sigmoid(f)
  float* go  = gf + (size_t)BTD;             // [B,T,D] sigmoid(o)
  float* chA = go + (size_t)BTD;             // [B,NCH,D] chunk forget products
  float* chB = chA + (size_t)(B_ * NCH * D_);// [B,NCH,D] chunk offsets
  float* h   = (float*)d_out;

  const size_t smem_bytes = (size_t)(XROWS * XSTR + WELEMS) * sizeof(float); // ~232 KB (<320 KB WGP LDS)
  (void)hipFuncSetAttribute((const void*)qrnn_gates_wmma,
                            hipFuncAttributeMaxDynamicSharedMemorySize, (int)smem_bytes);

  qrnn_gates_wmma<<<dim3(B_ * T_ / 128, D_ / 16), 256, smem_bytes, stream>>>(
      x, Wz, Wf, Wo, bz, bf, bo, gz, gf, go);
  qrnn_chunk_reduce<<<dim3(B_ * NCH), 256, 0, stream>>>(gz, gf, chA, chB);
  qrnn_chunk_apply<<<dim3(B_ * NCH), 256, 0, stream>>>(gz, gf, go, chA, chB, h);
}